// EGNN_Layer_51496657879383
// MI455X (gfx1250) — compile-verified
//
#include <hip/hip_runtime.h>
#include <hip/hip_bf16.h>

typedef __attribute__((ext_vector_type(2))) float v2f;
typedef __attribute__((ext_vector_type(8))) float v8f;

#define NODES 50000
#define EDGES 640000
#define DIM   128
#define STG_LD 132   // padded stride for 16x128 staging tiles (bank-conflict free)

__device__ __forceinline__ v8f wmma_f32(v2f a, v2f b, v8f c) {
  // D = A(16x4 f32) * B(4x16 f32) + C(16x16 f32)
  return __builtin_amdgcn_wmma_f32_16x16x4_f32(false, a, false, b, (short)0, c,
                                               false, false);
}

__global__ void egnn_zero_kernel(float4* __restrict__ p, int n4) {
  int i = blockIdx.x * blockDim.x + threadIdx.x;
  if (i < n4) p[i] = make_float4(0.f, 0.f, 0.f, 0.f);
}

// ---------------------------------------------------------------------------
// Edge kernel: one wave == one 16-edge tile.
//   m_ij = relu([h_row | h_col | d2] @ We1 + be1) @ We2 + be2
//   scale = tanh(relu(m_ij @ Wc + bc))
//   atomic: delta[row] += rel_pos*scale ; agg[row] += m_ij
// Weights live in LDS in a K-swizzled layout sW[kb][n][kk] (kk = k & 3) so a
// lane's B fragment {K=4kb+koff, K=4kb+koff+1} is one aligned ds_load_b64.
// ---------------------------------------------------------------------------
__global__ __launch_bounds__(256) void egnn_edge_kernel(
    const float* __restrict__ h, const float* __restrict__ x,
    const int* __restrict__ eidx,
    const float* __restrict__ We1, const float* __restrict__ be1,
    const float* __restrict__ We2, const float* __restrict__ be2,
    const float* __restrict__ Wc,  const float* __restrict__ bc,
    float* __restrict__ agg, float* __restrict__ delta)
{
  __shared__ float sWe1s[64 * 512];          // 131072 B : We1[0:256,:] swizzled
  __shared__ float sWe2s[32 * 512];          //  65536 B : We2 swizzled
  __shared__ float sW256[DIM];               // We1[256,:] (dist2 row)
  __shared__ float sWc[DIM];
  __shared__ float sbe1[DIM];
  __shared__ float sbe2[DIM];
  __shared__ float sStage[8][16 * STG_LD];   //  67584 B
  __shared__ float sD2[8][16];
  __shared__ int   sRow[8][16];

  const int tid = threadIdx.x;
  // stage in LDS-linear order (conflict-free stores; strided L2 reads)
  for (int j = tid; j < 64 * 512; j += 256) {
    const int kb = j >> 9, n = (j >> 2) & 127, kk = j & 3;
    sWe1s[j] = We1[(size_t)(kb * 4 + kk) * DIM + n];
  }
  for (int j = tid; j < 32 * 512; j += 256) {
    const int kb = j >> 9, n = (j >> 2) & 127, kk = j & 3;
    sWe2s[j] = We2[(size_t)(kb * 4 + kk) * DIM + n];
  }
  if (tid < DIM) {
    sW256[tid] = We1[256 * DIM + tid];
    sWc[tid] = Wc[tid]; sbe1[tid] = be1[tid]; sbe2[tid] = be2[tid];
  }

  const int w    = tid >> 5;
  const int lane = tid & 31;
  const int e16  = lane & 15;          // edge-in-tile / N-column lane id
  const int koff = (lane >> 4) * 2;    // K sub-offset for A/B fragments
  const int hi8  = (lane >> 4) * 8;    // M offset for C fragments
  const int ebase = (blockIdx.x * 8 + w) * 16;   // E == 16*8*gridDim exactly

  // per-edge scalars (upper half lanes replicate lower half)
  const int r = eidx[ebase + e16];
  const int c = eidx[EDGES + ebase + e16];
  const float dx = x[r * 3 + 0] - x[c * 3 + 0];
  const float dy = x[r * 3 + 1] - x[c * 3 + 1];
  const float dz = x[r * 3 + 2] - x[c * 3 + 2];
  const float d2 = dx * dx + dy * dy + dz * dz;
  if (lane < 16) { sD2[w][lane] = d2; sRow[w][lane] = r; }

  __syncthreads();

  // ---- GEMM1: A[16x256] = [h_row | h_col], B = We1[0:256,:], + d2 rank-1 ----
  v8f acc[8] = {};
  const float* hr = h + (size_t)r * DIM;
  const float* hc = h + (size_t)c * DIM;
  for (int kb = 0; kb < 64; ++kb) {
    const int k = kb * 4 + koff;
    const float* asrc = (k < DIM) ? (hr + k) : (hc + (k - DIM));
    const v2f a = *(const v2f*)asrc;                     // global_load_b64
    const float* bbase = &sWe1s[kb * 512 + koff];
    #pragma unroll
    for (int t = 0; t < 8; ++t) {
      const int n = t * 16 + e16;
      const v2f b = *(const v2f*)(bbase + n * 4);        // ds_load_b64
      acc[t] = wmma_f32(a, b, acc[t]);
    }
  }
  // epilogue: + be1 + d2 * We1[256,:], ReLU -> LDS staging (row-major 16x132)
  #pragma unroll
  for (int t = 0; t < 8; ++t) {
    const int n = t * 16 + e16;
    const float w256 = sW256[n];
    const float bias = sbe1[n];
    #pragma unroll
    for (int v = 0; v < 8; ++v) {
      const int m = v + hi8;
      const float val = acc[t][v] + bias + sD2[w][m] * w256;
      sStage[w][m * STG_LD + n] = fmaxf(val, 0.0f);
    }
  }
  __syncthreads();

  // ---- GEMM2: m2[16x128] = stage[16x128] @ We2 ----
  v8f acc2[8] = {};
  for (int kb = 0; kb < 32; ++kb) {
    const int k = kb * 4 + koff;
    const v2f a = *(const v2f*)&sStage[w][e16 * STG_LD + k];  // ds_load_b64
    const float* bbase = &sWe2s[kb * 512 + koff];
    #pragma unroll
    for (int t = 0; t < 8; ++t) {
      const int n = t * 16 + e16;
      const v2f b = *(const v2f*)(bbase + n * 4);
      acc2[t] = wmma_f32(a, b, acc2[t]);
    }
  }
  __syncthreads();                     // everyone done reading stage
  #pragma unroll
  for (int t = 0; t < 8; ++t) {
    const int n = t * 16 + e16;
    const float bias = sbe2[n];
    #pragma unroll
    for (int v = 0; v < 8; ++v) {
      const int m = v + hi8;
      sStage[w][m * STG_LD + n] = acc2[t][v] + bias;   // m_ij
    }
  }
  __syncthreads();

  // ---- coord gate + delta atomics (lane e owns edge e) ----
  if (lane < 16) {
    float s = bc[0];
    for (int k = 0; k < DIM; ++k) s += sStage[w][lane * STG_LD + k] * sWc[k];
    const float sc = tanhf(fmaxf(s, 0.0f));
    atomicAdd(&delta[(size_t)r * 3 + 0], dx * sc);
    atomicAdd(&delta[(size_t)r * 3 + 1], dy * sc);
    atomicAdd(&delta[(size_t)r * 3 + 2], dz * sc);
  }

  // ---- agg segment-sum: each lane owns 4 columns across 16 edges ----
  const int n0 = lane * 4;
  for (int e = 0; e < 16; ++e) {
    const int re = sRow[w][e];
    float* dst = agg + (size_t)re * DIM + n0;
    #pragma unroll
    for (int j = 0; j < 4; ++j)
      atomicAdd(dst + j, sStage[w][e * STG_LD + n0 + j]);
  }
}

// ---------------------------------------------------------------------------
// Node kernel: one wave == one 16-node tile.
//   h_new = relu([h | agg] @ Wn + bn) ; h_out = LN(h + h_new)
//   x_new = x + crw * delta
// ---------------------------------------------------------------------------
__global__ __launch_bounds__(256) void egnn_node_kernel(
    const float* __restrict__ h, const float* __restrict__ x,
    const float* __restrict__ Wn, const float* __restrict__ bn,
    const float* __restrict__ lng, const float* __restrict__ lnb,
    const float* __restrict__ crw,
    const float* __restrict__ agg, const float* __restrict__ delta,
    float* __restrict__ hout, float* __restrict__ xout)
{
  __shared__ float sWns[64 * 512];           // 131072 B : Wn swizzled
  __shared__ float sbn[DIM];
  __shared__ float sg[DIM];
  __shared__ float sb[DIM];
  __shared__ float sStage[8][16 * STG_LD];   //  67584 B
  __shared__ float sMu[8][16];
  __shared__ float sRs[8][16];

  const int tid = threadIdx.x;
  for (int j = tid; j < 64 * 512; j += 256) {
    const int kb = j >> 9, n = (j >> 2) & 127, kk = j & 3;
    sWns[j] = Wn[(size_t)(kb * 4 + kk) * DIM + n];
  }
  if (tid < DIM) { sbn[tid] = bn[tid]; sg[tid] = lng[tid]; sb[tid] = lnb[tid]; }
  __syncthreads();

  const int w    = tid >> 5;
  const int lane = tid & 31;
  const int e16  = lane & 15;
  const int koff = (lane >> 4) * 2;
  const int hi8  = (lane >> 4) * 8;
  const int nbase = (blockIdx.x * 8 + w) * 16;
  const bool valid = (nbase < NODES);        // NODES % 16 == 0 -> whole tile valid
  const int node = valid ? (nbase + e16) : 0;

  // ---- GEMM: A[16x256] = [h | agg], B = Wn ----
  v8f acc[8] = {};
  const float* ha = h   + (size_t)node * DIM;
  const float* ga = agg + (size_t)node * DIM;
  for (int kb = 0; kb < 64; ++kb) {
    const int k = kb * 4 + koff;
    const float* asrc = (k < DIM) ? (ha + k) : (ga + (k - DIM));
    const v2f a = *(const v2f*)asrc;
    const float* bbase = &sWns[kb * 512 + koff];
    #pragma unroll
    for (int t = 0; t < 8; ++t) {
      const int n = t * 16 + e16;
      const v2f b = *(const v2f*)(bbase + n * 4);
      acc[t] = wmma_f32(a, b, acc[t]);
    }
  }
  // y = h + relu(acc + bn) -> staging
  #pragma unroll
  for (int t = 0; t < 8; ++t) {
    const int n = t * 16 + e16;
    const float bias = sbn[n];
    #pragma unroll
    for (int v = 0; v < 8; ++v) {
      const int m  = v + hi8;
      const int nd = valid ? (nbase + m) : 0;
      const float hv = h[(size_t)nd * DIM + n];
      sStage[w][m * STG_LD + n] = hv + fmaxf(acc[t][v] + bias, 0.0f);
    }
  }
  __syncthreads();

  if (lane < 16) {
    float mu = 0.0f;
    for (int k = 0; k < DIM; ++k) mu += sStage[w][lane * STG_LD + k];
    mu *= (1.0f / DIM);
    float var = 0.0f;
    for (int k = 0; k < DIM; ++k) {
      const float d = sStage[w][lane * STG_LD + k] - mu;
      var += d * d;
    }
    var *= (1.0f / DIM);
    sMu[w][lane] = mu;
    sRs[w][lane] = rsqrtf(var + 1e-5f);
  }
  __syncthreads();

  if (valid) {
    const int n0 = lane * 4;
    for (int m = 0; m < 16; ++m) {
      const float mu = sMu[w][m], rs = sRs[w][m];
      const size_t o = (size_t)(nbase + m) * DIM + n0;
      #pragma unroll
      for (int j = 0; j < 4; ++j)
        hout[o + j] = (sStage[w][m * STG_LD + n0 + j] - mu) * rs * sg[n0 + j] + sb[n0 + j];
    }
    if (lane < 16) {
      const float cw = crw[0];
      const size_t o = (size_t)(nbase + lane) * 3;
      #pragma unroll
      for (int i = 0; i < 3; ++i)
        xout[o + i] = x[o + i] + cw * delta[o + i];
    }
  }
}

extern "C" void kernel_launch(void* const* d_in, const int* in_sizes, int n_in,
                              void* d_out, int out_size, void* d_ws, size_t ws_size,
                              hipStream_t stream) {
  const float* h   = (const float*)d_in[0];
  const float* x   = (const float*)d_in[1];
  const int*   ei  = (const int*)d_in[2];
  const float* We1 = (const float*)d_in[3];
  const float* be1 = (const float*)d_in[4];
  const float* We2 = (const float*)d_in[5];
  const float* be2 = (const float*)d_in[6];
  const float* Wc  = (const float*)d_in[7];
  const float* bc  = (const float*)d_in[8];
  const float* Wn  = (const float*)d_in[9];
  const float* bn  = (const float*)d_in[10];
  const float* lg  = (const float*)d_in[11];
  const float* lb  = (const float*)d_in[12];
  const float* crw = (const float*)d_in[13];

  float* agg   = (float*)d_ws;                     // [N,128]
  float* delta = agg + (size_t)NODES * DIM;        // [N,3] (contiguous after agg)
  float* hout  = (float*)d_out;                    // [N,128]
  float* xout  = hout + (size_t)NODES * DIM;       // [N,3]

  const int zn4 = (NODES * DIM + NODES * 3) / 4;   // 6,550,000 % 4 == 0
  egnn_zero_kernel<<<(zn4 + 255) / 256, 256, 0, stream>>>((float4*)agg, zn4);

  // E = 640000 = 16 edges/tile * 8 waves/block * 5000 blocks (exact)
  egnn_edge_kernel<<<EDGES / (16 * 8), 256, 0, stream>>>(
      h, x, ei, We1, be1, We2, be2, Wc, bc, agg, delta);

  // N = 50000 -> 3125 tiles -> 391 blocks (last block partially idle)
  egnn_node_kernel<<<(NODES / 16 + 7) / 8, 256, 0, stream>>>(
      h, x, Wn, bn, lg, lb, crw, agg, delta, hout, xout);
}